// MultiLayerGNN_47150150975850
// MI455X (gfx1250) — compile-verified
//
#include <hip/hip_runtime.h>

typedef __attribute__((ext_vector_type(2))) float v2f;
typedef __attribute__((ext_vector_type(8))) float v8f;

// ---------------------------------------------------------------------------
// C[M x Ncols] = A[M x K] @ B[K x Ncols], all f32, using V_WMMA_F32_16X16X4_F32.
// One wave per 16x16 output tile. M, Ncols multiples of 16; K multiple of 4.
// A-layout (16x4): lanes 0-15 hold {K=k,k+1} for row M=lane; lanes 16-31 hold
// {K=k+2,k+3} for row M=lane-16. B-layout (4x16): mirrored with N across lanes.
// C/D layout: VGPR r -> row tileM*16 + r + 8*(lane>=16), col = tileN*16 + lane%16.
// ---------------------------------------------------------------------------
__global__ __launch_bounds__(256) void gemm_f32_wmma(
    const float* __restrict__ A, const float* __restrict__ B,
    float* __restrict__ C, int M, int K, int Ncols) {
  const int lane = threadIdx.x & 31;
  const int waveInBlock = threadIdx.x >> 5;
  const int wavesPerBlock = blockDim.x >> 5;
  const int tilesN = Ncols >> 4;
  const int totalTiles = (M >> 4) * tilesN;
  const int tile = blockIdx.x * wavesPerBlock + waveInBlock;
  if (tile >= totalTiles) return;  // wave-uniform: EXEC stays all-1s for WMMA

  const int tm = tile / tilesN;
  const int tn = tile - tm * tilesN;
  const int half = lane >> 4;   // 0: K pair {k,k+1}, 1: {k+2,k+3}
  const int lr = lane & 15;

  const float* arow = A + (size_t)(tm * 16 + lr) * K;
  const int col = tn * 16 + lr;

  v8f c = {0.f, 0.f, 0.f, 0.f, 0.f, 0.f, 0.f, 0.f};
  for (int k = 0; k < K; k += 4) {
    const int kk = k + 2 * half;
    v2f a = *(const v2f*)(arow + kk);          // 8B-aligned: kk even, row 256B-aligned
    v2f b;
    b.x = B[(size_t)kk * Ncols + col];
    b.y = B[(size_t)(kk + 1) * Ncols + col];
    // (neg_a, A, neg_b, B, c_mod, C, reuse_a, reuse_b)
    c = __builtin_amdgcn_wmma_f32_16x16x4_f32(false, a, false, b, (short)0, c,
                                              false, false);
  }

  float* crow = C + (size_t)(tm * 16 + 8 * half) * Ncols + col;
#pragma unroll
  for (int r = 0; r < 8; ++r) crow[(size_t)r * Ncols] = c[r];
}

// ---------------------------------------------------------------------------
// out[i] = bias[i & dmask]   (accumulator init with broadcast bias)
// ---------------------------------------------------------------------------
__global__ __launch_bounds__(256) void fill_bias(
    float* __restrict__ out, const float* __restrict__ bias,
    long long total, int dmask) {
  long long i = (long long)blockIdx.x * blockDim.x + threadIdx.x;
  if (i < total) out[i] = bias[(int)(i & dmask)];
}

// ---------------------------------------------------------------------------
// SPMM d=64: per edge e, out[dst[e], :] += w[e] * h[src[e], :]
// Each half-wave (16 lanes) owns one edge; lane lr handles float4 chunk lr.
// Gathers are coalesced 256B rows from L2-resident h; atomics hit L2.
// ---------------------------------------------------------------------------
__global__ __launch_bounds__(256) void spmm_d64(
    const int* __restrict__ src, const int* __restrict__ dst,
    const float* __restrict__ w, const float* __restrict__ h,
    float* __restrict__ out, long long E) {
  const int lane = threadIdx.x & 31;
  const int wv = threadIdx.x >> 5;
  const int half = lane >> 4, lr = lane & 15;
  const long long e =
      (long long)blockIdx.x * ((blockDim.x >> 5) * 2) + wv * 2 + half;
  if (e >= E) return;
  const int s = src[e];
  const int d = dst[e];
  const float ww = w[e];
  const float4 v = ((const float4*)(h + (size_t)s * 64))[lr];
  float* orow = out + (size_t)d * 64 + lr * 4;
  atomicAdd(orow + 0, v.x * ww);
  atomicAdd(orow + 1, v.y * ww);
  atomicAdd(orow + 2, v.z * ww);
  atomicAdd(orow + 3, v.w * ww);
}

// ---------------------------------------------------------------------------
// SPMM d=16: half-wave per edge, one lane per feature.
// ---------------------------------------------------------------------------
__global__ __launch_bounds__(256) void spmm_d16(
    const int* __restrict__ src, const int* __restrict__ dst,
    const float* __restrict__ w, const float* __restrict__ h,
    float* __restrict__ out, long long E) {
  const int lane = threadIdx.x & 31;
  const int wv = threadIdx.x >> 5;
  const int half = lane >> 4, f = lane & 15;
  const long long e =
      (long long)blockIdx.x * ((blockDim.x >> 5) * 2) + wv * 2 + half;
  if (e >= E) return;
  const int s = src[e];
  const int d = dst[e];
  const float ww = w[e];
  atomicAdd(out + (size_t)d * 16 + f, h[(size_t)s * 16 + f] * ww);
}

// ---------------------------------------------------------------------------
// In-place ReLU.
// ---------------------------------------------------------------------------
__global__ __launch_bounds__(256) void relu_ip(float* __restrict__ x,
                                               long long total) {
  long long i = (long long)blockIdx.x * blockDim.x + threadIdx.x;
  if (i < total) x[i] = fmaxf(x[i], 0.f);
}

// ---------------------------------------------------------------------------
// In-place row log_softmax over 16 classes. Half-wave per row, one lane per
// class; max/sum reductions via width-16 xor shuffles (wave32).
// ---------------------------------------------------------------------------
__global__ __launch_bounds__(256) void log_softmax16(float* __restrict__ io,
                                                     int N) {
  const int lane = threadIdx.x & 31;
  const int wv = threadIdx.x >> 5;
  const int half = lane >> 4, f = lane & 15;
  const int row = blockIdx.x * ((blockDim.x >> 5) * 2) + wv * 2 + half;
  if (row >= N) return;
  const float v = io[(size_t)row * 16 + f];
  float m = v;
#pragma unroll
  for (int off = 8; off; off >>= 1) m = fmaxf(m, __shfl_xor(m, off, 16));
  float s = expf(v - m);
#pragma unroll
  for (int off = 8; off; off >>= 1) s += __shfl_xor(s, off, 16);
  io[(size_t)row * 16 + f] = v - m - logf(s);
}

// ---------------------------------------------------------------------------
// Orchestration. Input order: x, edge_src, edge_dst, edge_weight, W1, b1, W2, b2
// ---------------------------------------------------------------------------
extern "C" void kernel_launch(void* const* d_in, const int* in_sizes, int n_in,
                              void* d_out, int out_size, void* d_ws,
                              size_t ws_size, hipStream_t stream) {
  (void)n_in; (void)out_size; (void)ws_size;
  const float* x    = (const float*)d_in[0];
  const int*   esrc = (const int*)d_in[1];
  const int*   edst = (const int*)d_in[2];
  const float* ew   = (const float*)d_in[3];
  const float* W1   = (const float*)d_in[4];
  const float* b1   = (const float*)d_in[5];
  const float* W2   = (const float*)d_in[6];
  const float* b2   = (const float*)d_in[7];
  float* out = (float*)d_out;

  const int NFEAT = 128, NHID = 64, NCLASS = 16;
  const int N = in_sizes[0] / NFEAT;   // 100000 (multiple of 16)
  const long long E = in_sizes[1];     // 1600000

  float* h0 = (float*)d_ws;                 // [N,64] x@W1
  float* h1 = h0 + (size_t)N * NHID;        // [N,64] spmm accumulator
  float* h2 = h0;                           // [N,16] reuses h0 (dead after spmm1)

  const int wavesPerBlock = 8;  // 256 threads

  // 1) h0 = x @ W1   (WMMA f32 16x16x4)
  {
    int tiles = (N / 16) * (NHID / 16);
    int blocks = (tiles + wavesPerBlock - 1) / wavesPerBlock;
    gemm_f32_wmma<<<blocks, 256, 0, stream>>>(x, W1, h0, N, NFEAT, NHID);
  }
  // 2) h1 = broadcast(b1)
  {
    long long total = (long long)N * NHID;
    fill_bias<<<(int)((total + 255) / 256), 256, 0, stream>>>(h1, b1, total,
                                                              NHID - 1);
  }
  // 3) h1 += A @ h0  (scatter-add)
  {
    long long edgesPerBlock = wavesPerBlock * 2;
    int blocks = (int)((E + edgesPerBlock - 1) / edgesPerBlock);
    spmm_d64<<<blocks, 256, 0, stream>>>(esrc, edst, ew, h0, h1, E);
  }
  // 4) h1 = relu(h1)
  {
    long long total = (long long)N * NHID;
    relu_ip<<<(int)((total + 255) / 256), 256, 0, stream>>>(h1, total);
  }
  // 5) h2 = h1 @ W2   (WMMA f32 16x16x4)
  {
    int tiles = (N / 16) * (NCLASS / 16);
    int blocks = (tiles + wavesPerBlock - 1) / wavesPerBlock;
    gemm_f32_wmma<<<blocks, 256, 0, stream>>>(h1, W2, h2, N, NHID, NCLASS);
  }
  // 6) out = broadcast(b2)
  {
    long long total = (long long)N * NCLASS;
    fill_bias<<<(int)((total + 255) / 256), 256, 0, stream>>>(out, b2, total,
                                                              NCLASS - 1);
  }
  // 7) out += A @ h2  (scatter-add)
  {
    long long edgesPerBlock = wavesPerBlock * 2;
    int blocks = (int)((E + edgesPerBlock - 1) / edgesPerBlock);
    spmm_d16<<<blocks, 256, 0, stream>>>(esrc, edst, ew, h2, out, E);
  }
  // 8) out = log_softmax(out) row-wise, in place
  {
    int rowsPerBlock = wavesPerBlock * 2;
    int blocks = (N + rowsPerBlock - 1) / rowsPerBlock;
    log_softmax16<<<blocks, 256, 0, stream>>>(out, N);
  }
}